// CombinedHiddenGCAEEncoder_16286515987226
// MI455X (gfx1250) — compile-verified
//
#include <hip/hip_runtime.h>
#include <math.h>

#define N_NODES 50000
#define N_EDGES 800000

typedef __attribute__((ext_vector_type(2))) float v2f_t;
typedef __attribute__((ext_vector_type(8))) float v8f_t;

// ---------------- utility kernels ----------------

__global__ void fill_f32(float* p, float v, int n) {
    int i = blockIdx.x * blockDim.x + threadIdx.x;
    if (i < n) p[i] = v;
}

__global__ void deg_edges(const long long* __restrict__ dst, float* __restrict__ deg, int e) {
    int i = blockIdx.x * blockDim.x + threadIdx.x;
    if (i < e) unsafeAtomicAdd(&deg[dst[i]], 1.0f);
}

__global__ void deg_to_dinv(float* __restrict__ d, int n) {
    int i = blockIdx.x * blockDim.x + threadIdx.x;
    if (i < n) {
        float v = d[i];
        d[i] = (v > 0.0f) ? rsqrtf(v) : 0.0f;
    }
}

// ---------------- WMMA fp32 GEMM: H[n x NCOLS] = X[n x 64] @ W[64 x NCOLS] ----------------
// One wave computes a full 16-row x NCOLS output tile using NCT 16x16x4 fp32 WMMAs per K-step.
// Early exit is wave-uniform so EXEC is all-ones for every WMMA (ISA requirement).

template<int NCT>
__global__ void gcn_gemm_wmma(const float* __restrict__ X, const float* __restrict__ W,
                              float* __restrict__ H, int nRowTiles) {
    constexpr int NCOLS = NCT * 16;
    int wave = (int)((blockIdx.x * blockDim.x + threadIdx.x) >> 5);
    if (wave >= nRowTiles) return;                 // uniform per wave
    int lane = threadIdx.x & 31;
    int r    = lane & 15;                          // row within tile (A) / col (B,C,D)
    int kh   = lane >> 4;                          // K-half selector

    const float* xrow = X + (size_t)(wave * 16 + r) * 64;

    v8f_t acc[NCT] = {};

    #pragma unroll
    for (int k = 0; k < 64; k += 4) {
        // A fragment: lane holds A[r, k + kh*2 + j] in vgpr j
        float2 av = *(const float2*)(xrow + k + kh * 2);
        v2f_t a; a.x = av.x; a.y = av.y;
        #pragma unroll
        for (int t = 0; t < NCT; ++t) {
            // B fragment: lane holds B[k + kh*2 + j, t*16 + r] in vgpr j
            v2f_t b;
            b.x = W[(k + kh * 2 + 0) * NCOLS + t * 16 + r];
            b.y = W[(k + kh * 2 + 1) * NCOLS + t * 16 + r];
            acc[t] = __builtin_amdgcn_wmma_f32_16x16x4_f32(
                false, a, false, b, (short)0, acc[t], false, false);
        }
    }

    // C/D layout: vgpr i, lanes 0-15 -> M=i, lanes 16-31 -> M=8+i
    #pragma unroll
    for (int t = 0; t < NCT; ++t) {
        #pragma unroll
        for (int i = 0; i < 8; ++i) {
            H[(size_t)(wave * 16 + kh * 8 + i) * NCOLS + t * 16 + r] = acc[t][i];
        }
    }
}

// ---------------- edge-parallel aggregation: acc[dst] += h[src] * dinv[src]*dinv[dst] ----------------
// One wave per edge; lane covers 2 features (D=64) or 1 feature (D=32).
// Edge list + node features fit in the 192MB L2, so gathers and f32 atomics resolve in L2.

template<int D>
__global__ void gcn_aggregate(const float* __restrict__ h,
                              const long long* __restrict__ src,
                              const long long* __restrict__ dst,
                              const float* __restrict__ dinv,
                              float* __restrict__ acc, int e) {
    int wave = (int)((blockIdx.x * blockDim.x + threadIdx.x) >> 5);
    if (wave >= e) return;
    int lane = threadIdx.x & 31;
    long long s = src[wave];
    long long d = dst[wave];
    float nrm = dinv[s] * dinv[d];
    if (D == 64) {
        float2 v = *(const float2*)(h + (size_t)s * 64 + lane * 2);
        unsafeAtomicAdd(&acc[(size_t)d * 64 + lane * 2 + 0], v.x * nrm);
        unsafeAtomicAdd(&acc[(size_t)d * 64 + lane * 2 + 1], v.y * nrm);
    } else {
        float v = h[(size_t)s * 32 + lane];
        unsafeAtomicAdd(&acc[(size_t)d * 32 + lane], v * nrm);
    }
}

// ---------------- finalize: add self-loop term + bias, optional tanh ----------------

template<bool TANH, int D>
__global__ void gcn_finalize(const float* __restrict__ acc, const float* __restrict__ h,
                             const float* __restrict__ dinv, const float* __restrict__ b,
                             float* __restrict__ y, int n) {
    int i = blockIdx.x * blockDim.x + threadIdx.x;
    if (i >= n * D) return;
    int node = i / D;
    int f    = i % D;
    float di = dinv[node];
    float v  = acc[i] + h[i] * di * di + b[f];
    y[i] = TANH ? tanhf(v) : v;
}

// ---------------- launcher ----------------

extern "C" void kernel_launch(void* const* d_in, const int* in_sizes, int n_in,
                              void* d_out, int out_size, void* d_ws, size_t ws_size,
                              hipStream_t stream) {
    (void)in_sizes; (void)n_in; (void)out_size; (void)ws_size;

    const float*      x   = (const float*)d_in[0];
    const long long*  ei  = (const long long*)d_in[1];   // int64, shape (2, E)
    const float*      W1  = (const float*)d_in[2];
    const float*      b1  = (const float*)d_in[3];
    const float*      W2  = (const float*)d_in[4];
    const float*      b2  = (const float*)d_in[5];
    const float*      W3  = (const float*)d_in[6];
    const float*      b3  = (const float*)d_in[7];
    float*            out = (float*)d_out;

    const long long* src = ei;
    const long long* dst = ei + N_EDGES;

    const int N = N_NODES, E = N_EDGES;

    // workspace layout (floats): dinv[N] | h[N*64] | acc[N*64] | y[N*64]
    float* dinv = (float*)d_ws;
    float* h    = dinv + N;
    float* acc  = h + (size_t)N * 64;
    float* y    = acc + (size_t)N * 64;

    const int TPB = 256;
    const int rowTiles   = N / 16;                       // 3125 (N divisible by 16)
    const int gemmBlocks = (rowTiles + 7) / 8;           // 8 waves per 256-thread block
    const int edgeBlocks = (E + 7) / 8;                  // one wave per edge
    const int nodeBlocks = (N + TPB - 1) / TPB;
    const int nd64Blocks = ((size_t)N * 64 + TPB - 1) / TPB;
    const int nd32Blocks = ((size_t)N * 32 + TPB - 1) / TPB;
    const int edgeThread = (E + TPB - 1) / TPB;

    // degrees over A + I: deg = 1 (self-loop) + atomic count on dst; then rsqrt
    fill_f32<<<nodeBlocks, TPB, 0, stream>>>(dinv, 1.0f, N);
    deg_edges<<<edgeThread, TPB, 0, stream>>>(dst, dinv, E);
    deg_to_dinv<<<nodeBlocks, TPB, 0, stream>>>(dinv, N);

    // ---- layer 1: x @ W1 -> h ; aggregate ; tanh -> y
    gcn_gemm_wmma<4><<<gemmBlocks, TPB, 0, stream>>>(x, W1, h, rowTiles);
    fill_f32<<<nd64Blocks, TPB, 0, stream>>>(acc, 0.0f, N * 64);
    gcn_aggregate<64><<<edgeBlocks, TPB, 0, stream>>>(h, src, dst, dinv, acc, E);
    gcn_finalize<true, 64><<<nd64Blocks, TPB, 0, stream>>>(acc, h, dinv, b1, y, N);

    // ---- layer 2: y @ W2 -> h ; aggregate ; tanh -> y (in place: finalize reads acc/h only)
    gcn_gemm_wmma<4><<<gemmBlocks, TPB, 0, stream>>>(y, W2, h, rowTiles);
    fill_f32<<<nd64Blocks, TPB, 0, stream>>>(acc, 0.0f, N * 64);
    gcn_aggregate<64><<<edgeBlocks, TPB, 0, stream>>>(h, src, dst, dinv, acc, E);
    gcn_finalize<true, 64><<<nd64Blocks, TPB, 0, stream>>>(acc, h, dinv, b2, y, N);

    // ---- layer 3: y @ W3 -> h (N x 32) ; aggregate ; no tanh -> out
    gcn_gemm_wmma<2><<<gemmBlocks, TPB, 0, stream>>>(y, W3, h, rowTiles);
    fill_f32<<<nd32Blocks, TPB, 0, stream>>>(acc, 0.0f, N * 32);
    gcn_aggregate<32><<<edgeBlocks, TPB, 0, stream>>>(h, src, dst, dinv, acc, E);
    gcn_finalize<false, 32><<<nd32Blocks, TPB, 0, stream>>>(acc, h, dinv, b3, out, N);
}